// MSDeformAttn_88957362635401
// MI455X (gfx1250) — compile-verified
//
#include <hip/hip_runtime.h>
#include <hip/hip_bf16.h>

// ---------------------------------------------------------------------------
// MS Deformable Attention, fp32 end-to-end, CDNA5 (gfx1250) WMMA f32 path.
// ---------------------------------------------------------------------------

typedef __attribute__((ext_vector_type(2))) float v2f;
typedef __attribute__((ext_vector_type(8))) float v8f;

#define B_   8
#define Q_   300
#define C_   256
#define H_   8
#define L_   4
#define K_   4
#define D_   32
#define N_   21760          // 128^2 + 64^2 + 32^2 + 16^2
#define SCALE_ 8.0f

// ---- Generic fp32 WMMA GEMM: C[M,N] = A[M,K] * B[K,N] + bias[N] ----------
// Block: 256 threads = 8 waves. Block tile 64x64. Wave tile 16x32
// (two 16x16 accumulators sharing one A fragment).
// Uses V_WMMA_F32_16X16X4_F32 (full fp32, matches reference precision).
//
// Bs uses a k-pair interleaved layout: element (k,n) lives at
//   Bs[(k>>1)*LDBP + 2*n + (k&1)]
// so a B fragment {B[ka][n], B[ka+1][n]} (ka even) is one aligned 8-byte
// LDS read straight into a consecutive VGPR pair (legal WMMA operand,
// no register shuffles). LDBP is a multiple of 32 so the two lane-halves
// (pair rows P / P+1) map to disjoint bank ranges.

#define BM 64
#define BN 64
#define BK 32
#define LDA  36    // BM x BK A tile, padded; rows stay 16B-aligned
#define LDBP 160   // (BK/2) k-pair rows of 2*BN floats, padded to 32-multiple

__global__ __launch_bounds__(256)
void wmma_gemm_bias_f32(const float* __restrict__ A,
                        const float* __restrict__ B,
                        const float* __restrict__ bias,
                        float* __restrict__ C,
                        int M, int N, int K)
{
    __shared__ float As[BM * LDA];
    __shared__ float Bs[(BK / 2) * LDBP];

    const int tid  = threadIdx.x;
    const int lane = tid & 31;
    const int wave = tid >> 5;       // 0..7
    const int wrow = wave & 3;       // 4 wave-rows of 16
    const int wcol = wave >> 2;      // 2 wave-cols of 32

    const int rowBase = blockIdx.x * BM;
    const int colBase = blockIdx.y * BN;

    const int m  = lane & 15;        // row/col within 16x16 fragment
    const int kh = lane >> 4;        // lane-half selector

    v8f acc0 = {};
    v8f acc1 = {};

    for (int k0 = 0; k0 < K; k0 += BK) {
        // ---- stage A tile (64x32 floats = 512 float4, 2 per thread) ----
        #pragma unroll
        for (int p = 0; p < 2; ++p) {
            int idx = tid + p * 256;         // float4 index 0..511
            int r   = idx >> 3;              // 8 float4 per 32-float row
            int c4  = idx & 7;
            int gr  = rowBase + r;
            int grc = gr < M ? gr : (M - 1);
            const float* ga = A + (size_t)grc * K + k0 + c4 * 4;
            float4 v = *(const float4*)ga;
            if (gr >= M) { v.x = 0.f; v.y = 0.f; v.z = 0.f; v.w = 0.f; }
            *(float4*)(&As[r * LDA + c4 * 4]) = v;
            if (k0 + BK < K) __builtin_prefetch(ga + BK, 0, 1);  // next chunk
        }
        // ---- stage B tile (32x64 floats), k-pair interleaved ----
        #pragma unroll
        for (int p = 0; p < 2; ++p) {
            int idx = tid + p * 256;         // float4 index 0..511
            int r   = idx >> 4;              // k row (16 float4 per row)
            int c4  = idx & 15;
            const float* gb = B + (size_t)(k0 + r) * N + colBase + c4 * 4;
            float4 v = *(const float4*)gb;
            float* dst = &Bs[(r >> 1) * LDBP + (c4 * 4) * 2 + (r & 1)];
            dst[0] = v.x; dst[2] = v.y; dst[4] = v.z; dst[6] = v.w;
            if (k0 + BK < K) __builtin_prefetch(gb + (size_t)BK * N, 0, 1);
        }
        __syncthreads();

        // ---- 16 x wmma_f32_16x16x4 over the 32-deep K chunk ----
        #pragma unroll
        for (int kk = 0; kk < BK; kk += 4) {
            const int ka = kk + 2 * kh;      // always even
            v2f a;
            a.x = As[(wrow * 16 + m) * LDA + ka];
            a.y = As[(wrow * 16 + m) * LDA + ka + 1];
            const v2f b0 = *(const v2f*)(&Bs[(ka >> 1) * LDBP +
                                             (wcol * 32 + m) * 2]);
            const v2f b1 = *(const v2f*)(&Bs[(ka >> 1) * LDBP +
                                             (wcol * 32 + 16 + m) * 2]);
            acc0 = __builtin_amdgcn_wmma_f32_16x16x4_f32(
                       false, a, false, b0, (short)0, acc0, false, false);
            acc1 = __builtin_amdgcn_wmma_f32_16x16x4_f32(
                       false, a, false, b1, (short)0, acc1, false, false);
        }
        __syncthreads();
    }

    // ---- epilogue: vgpr r holds row (r + 8*kh), col (lane&15) ----
    const int col = colBase + wcol * 32 + m;
    const float bv0 = bias[col];
    const float bv1 = bias[col + 16];

    if (rowBase + BM <= M) {
        // full tile: unconditional coalesced stores
        #pragma unroll
        for (int r = 0; r < 8; ++r) {
            const int row = rowBase + wrow * 16 + r + 8 * kh;
            const size_t o = (size_t)row * N;
            C[o + col]      = acc0[r] + bv0;
            C[o + col + 16] = acc1[r] + bv1;
        }
    } else {
        #pragma unroll
        for (int r = 0; r < 8; ++r) {
            const int row = rowBase + wrow * 16 + r + 8 * kh;
            if (row < M) {
                const size_t o = (size_t)row * N;
                C[o + col]      = acc0[r] + bv0;
                C[o + col + 16] = acc1[r] + bv1;
            }
        }
    }
}

// ---- softmax over L*K=16 + sampling locations, 1 thread per (b,q,h) ------
__global__ __launch_bounds__(256)
void softmax_loc_kernel(const float* __restrict__ logits,   // (BQ,128)
                        const float* __restrict__ offsets,  // (BQ,256)
                        const float* __restrict__ refpts,   // (BQ,2)
                        float* __restrict__ attn,           // (BQ,128)
                        float* __restrict__ loc,            // (BQ,256)
                        int BQ)
{
    int t = blockIdx.x * blockDim.x + threadIdx.x;
    if (t >= BQ * H_) return;
    int bq = t >> 3;
    int h  = t & 7;

    const float* lg = logits + (size_t)bq * 128 + h * 16;
    float mx = -3.0e38f;
    #pragma unroll
    for (int i = 0; i < 16; ++i) mx = fmaxf(mx, lg[i]);
    float e[16], s = 0.f;
    #pragma unroll
    for (int i = 0; i < 16; ++i) { e[i] = __expf(lg[i] - mx); s += e[i]; }
    const float inv = 1.f / s;

    const float rx = refpts[bq * 2 + 0];
    const float ry = refpts[bq * 2 + 1];
    const float* off = offsets + (size_t)bq * 256 + h * 32;
    float* lo = loc + (size_t)bq * 256 + h * 32;
    float* at = attn + (size_t)bq * 128 + h * 16;
    #pragma unroll
    for (int i = 0; i < 16; ++i) {
        at[i] = e[i] * inv;
        lo[2 * i + 0] = rx + off[2 * i + 0] * (1.0f / SCALE_);
        lo[2 * i + 1] = ry + off[2 * i + 1] * (1.0f / SCALE_);
    }
}

// ---- bilinear sampling + weighted sum: 1 wave per (b,q,h), lane = channel -
__global__ __launch_bounds__(256)
void msda_sample_kernel(const float* __restrict__ value,   // (B,N,H*D)
                        const float* __restrict__ attn,    // (BQ,128)
                        const float* __restrict__ loc,     // (BQ,256)
                        float* __restrict__ results)       // (BQ,256)
{
    const int wave = blockIdx.x * (blockDim.x >> 5) + (threadIdx.x >> 5);
    const int lane = threadIdx.x & 31;                     // channel d
    const int total = B_ * Q_ * H_;
    if (wave >= total) return;

    const int h  = wave % H_;
    const int bq = wave / H_;
    const int b  = bq / Q_;

    const int Hls[L_]    = {128, 64, 32, 16};
    const int Wls[L_]    = {128, 64, 32, 16};
    const int starts[L_] = {0, 16384, 20480, 21504};

    const float* lo = loc  + (size_t)bq * 256 + h * 32;
    const float* at = attn + (size_t)bq * 128 + h * 16;

    float acc = 0.f;
    #pragma unroll
    for (int l = 0; l < L_; ++l) {
        const int Hl = Hls[l], Wl = Wls[l];
        const float* vbase =
            value + ((size_t)b * N_ + starts[l]) * (size_t)C_ + h * D_ + lane;
        #pragma unroll
        for (int k = 0; k < K_; ++k) {
            const float w  = at[l * 4 + k];
            const float x  = lo[l * 8 + k * 2 + 0] * Wl - 0.5f;
            const float y  = lo[l * 8 + k * 2 + 1] * Hl - 0.5f;
            const float x0f = floorf(x), y0f = floorf(y);
            const float dx = x - x0f, dy = y - y0f;
            const int x0 = (int)x0f, y0 = (int)y0f;
            // 4 taps; validity is wave-uniform (same x,y for all lanes)
            const int   xi[4] = {x0, x0 + 1, x0, x0 + 1};
            const int   yi[4] = {y0, y0, y0 + 1, y0 + 1};
            const float wt[4] = {(1.f - dx) * (1.f - dy), dx * (1.f - dy),
                                 (1.f - dx) * dy,         dx * dy};
            #pragma unroll
            for (int t = 0; t < 4; ++t) {
                if (xi[t] >= 0 && xi[t] < Wl && yi[t] >= 0 && yi[t] < Hl) {
                    acc += w * wt[t] *
                           vbase[(size_t)(yi[t] * Wl + xi[t]) * C_];
                }
            }
        }
    }
    results[(size_t)bq * 256 + h * D_ + lane] = acc;
}

// ---------------------------------------------------------------------------
extern "C" void kernel_launch(void* const* d_in, const int* in_sizes, int n_in,
                              void* d_out, int out_size, void* d_ws, size_t ws_size,
                              hipStream_t stream) {
    const float* query    = (const float*)d_in[0];   // (B,Q,C)
    const float* refpts   = (const float*)d_in[1];   // (B,Q,2)
    const float* in_flat  = (const float*)d_in[2];   // (B,N,C)
    // d_in[3]: spatial shapes (int32) — values are fixed, hardcoded above
    const float* W_value  = (const float*)d_in[4];   // (C,C)
    const float* b_value  = (const float*)d_in[5];   // (C)
    const float* W_attn   = (const float*)d_in[6];   // (C,128)
    const float* b_attn   = (const float*)d_in[7];   // (128)
    const float* W_off    = (const float*)d_in[8];   // (C,256)
    const float* b_off    = (const float*)d_in[9];   // (256)
    const float* W_out    = (const float*)d_in[10];  // (C,C)
    const float* b_out    = (const float*)d_in[11];  // (C)
    float* out = (float*)d_out;

    const int BQ = B_ * Q_;                 // 2400
    const int MV = B_ * N_;                 // 174080

    // workspace carve-up (floats)
    float* ws       = (float*)d_ws;
    float* value    = ws;                                   // MV*C
    float* offsets  = value   + (size_t)MV * C_;            // BQ*256
    float* logits   = offsets + (size_t)BQ * 256;           // BQ*128
    float* loc      = logits  + (size_t)BQ * 128;           // BQ*256
    float* results  = loc     + (size_t)BQ * 256;           // BQ*256

    dim3 blk(256);

    // 1) value = input_flatten @ W_value + b_value          (174080x256x256)
    wmma_gemm_bias_f32<<<dim3(MV / BM, C_ / BN), blk, 0, stream>>>(
        in_flat, W_value, b_value, value, MV, C_, C_);

    // 2) offsets = query @ W_off + b_off                    (2400x256x256)
    wmma_gemm_bias_f32<<<dim3((BQ + BM - 1) / BM, C_ / BN), blk, 0, stream>>>(
        query, W_off, b_off, offsets, BQ, C_, C_);

    // 3) logits = query @ W_attn + b_attn                   (2400x128x256)
    wmma_gemm_bias_f32<<<dim3((BQ + BM - 1) / BM, 128 / BN), blk, 0, stream>>>(
        query, W_attn, b_attn, logits, BQ, 128, C_);

    // 4) softmax over 16 points + sampling locations
    softmax_loc_kernel<<<(BQ * H_ + 255) / 256, blk, 0, stream>>>(
        logits, offsets, refpts, logits /*attn in-place*/, loc, BQ);

    // 5) bilinear sampling, one wave per (b,q,h)
    msda_sample_kernel<<<(BQ * H_ + 7) / 8, blk, 0, stream>>>(
        value, logits, loc, results);

    // 6) out = results @ W_out + b_out                      (2400x256x256)
    wmma_gemm_bias_f32<<<dim3((BQ + BM - 1) / BM, C_ / BN), blk, 0, stream>>>(
        results, W_out, b_out, out, BQ, C_, C_);
}